// MahalanobisMetricLoss_17738214933008
// MI455X (gfx1250) — compile-verified
//
#include <hip/hip_runtime.h>
#include <hip/hip_bf16.h>

typedef __attribute__((ext_vector_type(2))) float v2f;
typedef __attribute__((ext_vector_type(8))) float v8f;

#define BDIM 4096
#define DDIM 128
#define MARGIN_M 0.6f
#define EXTRA_E  0.04f

#define NBINS   (1 << 20)
#define DMAX    32.0f
#define BIN_SCL ((float)NBINS / DMAX)
#define CAND_CAP 8192

// ---- workspace layout (bytes) ----
#define OFF_D    0ull                                   // BDIM*BDIM f32 = 64 MB
#define OFF_HIST (OFF_D + (size_t)BDIM * BDIM * 4)      // NBINS u32    = 4 MB
#define OFF_SQN  (OFF_HIST + (size_t)NBINS * 4)         // BDIM f32
#define OFF_CAND (OFF_SQN + (size_t)BDIM * 4)           // CAND_CAP f32
#define OFF_SCAL (OFF_CAND + (size_t)CAND_CAP * 4)      // int scal[16]
#define OFF_ACC  (OFF_SCAL + 64)                        // float acc[16]

__device__ __forceinline__ float warp_sum(float v) {
#pragma unroll
  for (int o = 16; o > 0; o >>= 1) v += __shfl_xor(v, o, 32);
  return v;
}

// ---------- K0: per-row squared norms (1 wave per row) ----------
__global__ void k_sqnorm(const float* __restrict__ X, float* __restrict__ sqn) {
  int wave = threadIdx.x >> 5;
  int lane = threadIdx.x & 31;
  int row  = blockIdx.x * 8 + wave;
  const float4* p = (const float4*)(X + (size_t)row * DDIM);
  float4 v = p[lane];
  float s = v.x * v.x + v.y * v.y + v.z * v.z + v.w * v.w;
  s = warp_sum(s);
  if (lane == 0) sqn[row] = s;
}

// ---------- K1: WMMA pairwise distances (one 16x16 tile per wave) ----------
__global__ void k_pairdist(const float* __restrict__ X,
                           const float* __restrict__ sqn,
                           float* __restrict__ Dm) {
  int lane = threadIdx.x & 31;
  int wave = threadIdx.x >> 5;
  int tj = blockIdx.x;              // column tile [0,256)
  int ti = blockIdx.y * 8 + wave;   // row tile    [0,256)

  int mn = lane & 15;               // A row / B col handled by this lane
  int kb = (lane >> 4) * 2;         // K base for this half-wave (0 or 2)
  const float* arow = X + ((size_t)(ti * 16 + mn)) * DDIM + kb;
  const float* brow = X + ((size_t)(tj * 16 + mn)) * DDIM + kb;

  v8f c = {};
#pragma unroll
  for (int k = 0; k < DDIM; k += 4) {
    v2f a, b;
    a.x = arow[k]; a.y = arow[k + 1];   // A[m][kb+k], A[m][kb+k+1]
    b.x = brow[k]; b.y = brow[k + 1];   // B[kb+k][n], B[kb+k+1][n]
    c = __builtin_amdgcn_wmma_f32_16x16x4_f32(
        /*neg_a=*/false, a, /*neg_b=*/false, b,
        /*c_mod=*/(short)0, c, /*reuse_a=*/false, /*reuse_b=*/false);
  }

  int j = tj * 16 + mn;
  float snj = sqn[j];
  int mhalf = (lane >> 4) * 8;        // C/D: rows r (lanes 0-15), r+8 (lanes 16-31)
#pragma unroll
  for (int r = 0; r < 8; ++r) {
    int i = ti * 16 + mhalf + r;
    float sq = sqn[i] + snj - 2.0f * c[r];
    sq = sq > 0.0f ? sq : 0.0f;
    float d = sq > 0.0f ? sqrtf(sq) : 0.0f;
    Dm[(size_t)i * BDIM + j] = d;
  }
}

// ---------- K2: num_pairs from label counts ----------
__global__ void k_pairs(const int* __restrict__ T, int* __restrict__ scal) {
  __shared__ int cnt[16];
  if (threadIdx.x < 16) cnt[threadIdx.x] = 0;
  __syncthreads();
  for (int i = threadIdx.x; i < BDIM; i += blockDim.x)
    atomicAdd(&cnt[T[i] & 15], 1);
  __syncthreads();
  if (threadIdx.x == 0) {
    long long same = 0;
    for (int c = 0; c < 16; ++c) same += (long long)cnt[c] * (long long)cnt[c];
    scal[0] = (int)((same - BDIM) / 2);   // num_pairs
  }
}

// ---------- K3: histogram of negative-pair distances ----------
__global__ void k_hist(const float* __restrict__ Dm, const int* __restrict__ T,
                       unsigned* __restrict__ hist) {
  size_t total  = (size_t)BDIM * BDIM;
  size_t stride = (size_t)gridDim.x * blockDim.x;
  for (size_t idx = (size_t)blockIdx.x * blockDim.x + threadIdx.x;
       idx < total; idx += stride) {
    int i = (int)(idx >> 12);
    int j = (int)(idx & (BDIM - 1));
    if (T[i] != T[j]) {
      float d = Dm[idx];
      int b = (int)(d * BIN_SCL);
      b = b < (NBINS - 1) ? b : (NBINS - 1);
      atomicAdd(&hist[b], 1u);
    }
  }
}

// ---------- K4: find bin containing the k-th smallest negative distance ----------
__global__ void k_select_bin(const unsigned* __restrict__ hist,
                             int* __restrict__ scal) {
  __shared__ unsigned long long part[1024];
  const int CH = NBINS / 1024;
  int t = threadIdx.x;
  unsigned long long s = 0;
  for (int b = 0; b < CH; ++b) s += hist[(size_t)t * CH + b];
  part[t] = s;
  __syncthreads();
  if (t == 0) {
    unsigned long long k = (unsigned long long)scal[0];
    unsigned long long run = 0;
    int sel = 1023;
    for (int u = 0; u < 1024; ++u) {
      if (run + part[u] > k) { sel = u; break; }
      run += part[u];
    }
    int bin = sel * CH + CH - 1;
    for (int b = 0; b < CH; ++b) {
      unsigned h = hist[(size_t)sel * CH + b];
      if (run + h > k) { bin = sel * CH + b; break; }
      run += h;
    }
    scal[1] = bin;                 // bin containing order statistic
    scal[2] = (int)(k - run);      // rank within bin
  }
}

// ---------- K5: collect candidates in the selected bin ----------
__global__ void k_collect(const float* __restrict__ Dm, const int* __restrict__ T,
                          const int* __restrict__ scal,
                          float* __restrict__ cand, unsigned* __restrict__ count) {
  int bin = scal[1];
  size_t total  = (size_t)BDIM * BDIM;
  size_t stride = (size_t)gridDim.x * blockDim.x;
  for (size_t idx = (size_t)blockIdx.x * blockDim.x + threadIdx.x;
       idx < total; idx += stride) {
    int i = (int)(idx >> 12);
    int j = (int)(idx & (BDIM - 1));
    if (T[i] != T[j]) {
      float d = Dm[idx];
      int b = (int)(d * BIN_SCL);
      b = b < (NBINS - 1) ? b : (NBINS - 1);
      if (b == bin) {
        unsigned p = atomicAdd(count, 1u);
        if (p < CAND_CAP) cand[p] = d;
      }
    }
  }
}

// ---------- K6: exact rank-select among candidates ----------
__global__ void k_select_val(const float* __restrict__ cand,
                             const unsigned* __restrict__ count,
                             const int* __restrict__ scal,
                             float* __restrict__ thr) {
  int n = (int)*count; if (n > CAND_CAP) n = CAND_CAP;
  int want = scal[2];
  if (threadIdx.x == 0 && (n == 0 || want >= n)) {
    // fallback: bin midpoint (should not trigger for this data)
    *thr = ((float)scal[1] + 0.5f) / BIN_SCL;
  }
  for (int c = threadIdx.x; c < n; c += blockDim.x) {
    float v = cand[c];
    int rank = 0;
    for (int u = 0; u < n; ++u) {
      float w = cand[u];
      if (w < v || (w == v && u < c)) rank++;
    }
    if (rank == want) *thr = v;
  }
}

// ---------- K7: masked loss/accuracy reductions ----------
__global__ void k_loss_terms(const float* __restrict__ Dm, const int* __restrict__ T,
                             const float* __restrict__ thr, float* __restrict__ acc) {
  float thrv = *thr;
  float lpos = 0.0f, lneg = 0.0f, lright = 0.0f;
  size_t total  = (size_t)BDIM * BDIM;
  size_t stride = (size_t)gridDim.x * blockDim.x;
  for (size_t idx = (size_t)blockIdx.x * blockDim.x + threadIdx.x;
       idx < total; idx += stride) {
    int i = (int)(idx >> 12);
    int j = (int)(idx & (BDIM - 1));
    float d = Dm[idx];
    if (T[i] != T[j]) {
      if (d < MARGIN_M + EXTRA_E && d < thrv) lneg += (MARGIN_M + EXTRA_E) - d;
      if (d >= MARGIN_M) lright += 1.0f;
    } else {
      if (d > MARGIN_M - EXTRA_E) lpos += d - (MARGIN_M - EXTRA_E);
      if (d < MARGIN_M) lright += 1.0f;
    }
  }
  lpos = warp_sum(lpos); lneg = warp_sum(lneg); lright = warp_sum(lright);
  if ((threadIdx.x & 31) == 0) {
    atomicAdd(&acc[0], lpos);
    atomicAdd(&acc[1], lneg);
    atomicAdd(&acc[2], lright);
  }
}

// ---------- K8: finalize ----------
__global__ void k_final(const float* __restrict__ acc, const int* __restrict__ scal,
                        float* __restrict__ out) {
  if (threadIdx.x == 0) {
    float np = (float)scal[0];
    out[0] = (acc[0] + acc[1]) / (2.0f * np);
    out[1] = acc[2] / ((float)BDIM * (float)BDIM);
  }
}

extern "C" void kernel_launch(void* const* d_in, const int* in_sizes, int n_in,
                              void* d_out, int out_size, void* d_ws, size_t ws_size,
                              hipStream_t stream) {
  const float* X = (const float*)d_in[0];
  const int*   T = (const int*)d_in[1];
  float* out = (float*)d_out;

  char* ws = (char*)d_ws;
  float*    Dm    = (float*)(ws + OFF_D);
  unsigned* hist  = (unsigned*)(ws + OFF_HIST);
  float*    sqn   = (float*)(ws + OFF_SQN);
  float*    cand  = (float*)(ws + OFF_CAND);
  int*      scal  = (int*)(ws + OFF_SCAL);       // [0]=num_pairs [1]=bin [2]=rank [3]=count
  unsigned* count = (unsigned*)(scal + 3);
  float*    accf  = (float*)(ws + OFF_ACC);      // [0]=pos [1]=neg [2]=right [3]=threshold

  hipMemsetAsync(ws + OFF_HIST, 0, (size_t)NBINS * 4, stream);
  hipMemsetAsync(ws + OFF_SCAL, 0, 128, stream);

  k_sqnorm  <<<BDIM / 8, 256, 0, stream>>>(X, sqn);
  k_pairs   <<<1, 256, 0, stream>>>(T, scal);
  dim3 g1(BDIM / 16, BDIM / 128);
  k_pairdist<<<g1, 256, 0, stream>>>(X, sqn, Dm);
  k_hist    <<<8192, 256, 0, stream>>>(Dm, T, hist);
  k_select_bin<<<1, 1024, 0, stream>>>(hist, scal);
  k_collect <<<8192, 256, 0, stream>>>(Dm, T, scal, cand, count);
  k_select_val<<<1, 1024, 0, stream>>>(cand, count, scal, &accf[3]);
  k_loss_terms<<<8192, 256, 0, stream>>>(Dm, T, &accf[3], accf);
  k_final   <<<1, 32, 0, stream>>>(accf, scal, out);
}